// SWEncoder_57638461112690
// MI455X (gfx1250) — compile-verified
//
#include <hip/hip_runtime.h>
#include <hip/hip_bf16.h>

// Sliced-Wasserstein encoder, fused for MI455X (gfx1250, wave32).
//   xp = x @ proj  (fp32 via V_WMMA_F32_16X16X4_F32)
//   A staged global->LDS with GLOBAL_LOAD_ASYNC_TO_LDS_B128, double buffered
//   per-graph per-column sort (bitonic, LDS)
//   quantile gather + 1/256 scale
//
// Grid: (16 proj-tiles, 256 graphs), block = 256 threads = 8 waves.

typedef __attribute__((ext_vector_type(2))) float v2f;
typedef __attribute__((ext_vector_type(8))) float v8f;

#define N_NODES    131072
#define IN_CH      256
#define NUM_PROJ   256
#define NUM_QUANT  256
#define NUM_GRAPHS 256
#define NPG        512            // nodes per graph
#define KC         16             // K chunk staged in LDS
#define NCHUNK     (IN_CH / KC)   // 16
#define SA         20             // LDS stride of A chunk (16 + pad 4)
#define SB         20             // LDS stride of B (16 + pad 4)
#define SS         517            // LDS stride of sort columns (512 + pad 5)

// LDS layout (float indices):
//   pb  : [0, 256*SB)                  (proj tile, all K)
//   A0  : [A0OFF, A0OFF+512*SA)        (A chunk buf 0; reused as sort buf)
//   A1  : [A1OFF, A1OFF+512*SA)        (A chunk buf 1)
#define A0OFF  (256*SB)
#define A1OFF  (256*SB + 512*SA)
#define SBOFF  A0OFF              // sort buffer aliases A0 (16*SS=8272 <= 10240)
#define LDS_FLOATS (256*SB + 2*512*SA)

__device__ __forceinline__ void wait_asynccnt_8() {
#if __has_builtin(__builtin_amdgcn_s_wait_asynccnt)
    __builtin_amdgcn_s_wait_asynccnt(8);
#else
    asm volatile("s_wait_asynccnt 0x8" ::: "memory");
#endif
}
__device__ __forceinline__ void wait_asynccnt_0() {
#if __has_builtin(__builtin_amdgcn_s_wait_asynccnt)
    __builtin_amdgcn_s_wait_asynccnt(0);
#else
    asm volatile("s_wait_asynccnt 0x0" ::: "memory");
#endif
}

__global__ __launch_bounds__(256) void swe_fused_kernel(
    const float* __restrict__ x,          // (N_NODES, IN_CH)
    const float* __restrict__ proj,       // (IN_CH, NUM_PROJ)
    const float* __restrict__ cw,         // (NUM_QUANT)
    float* __restrict__ out)              // (NUM_GRAPHS, NUM_QUANT*NUM_PROJ)
{
    __shared__ float lds[LDS_FLOATS];

    const int tid  = threadIdx.x;
    const int pt   = blockIdx.x;          // projection tile: cols pt*16 .. pt*16+15
    const int g    = blockIdx.y;          // graph
    const int wave = tid >> 5;
    const int lane = tid & 31;
    const int half = lane >> 4;           // 0: lanes 0-15, 1: lanes 16-31
    const int lr   = lane & 15;
    const int m0   = wave * 64;           // this wave's 64 rows

    // 32-bit LDS byte offsets of the A buffers (flat addr low 32 bits == LDS offset)
    const unsigned aoff0 = (unsigned)(size_t)&lds[A0OFF];
    const unsigned aoff1 = (unsigned)(size_t)&lds[A1OFF];
    const unsigned long long xbase =
        (unsigned long long)(const void*)(x + (size_t)g * NPG * IN_CH);

    // ---- stage B (proj tile, full K=256) into LDS: pb[k*SB + c] ----
    {
        const int k = tid;                // 256 threads == 256 k values
        const float* src = proj + k * NUM_PROJ + pt * 16;
        #pragma unroll
        for (int i = 0; i < 4; ++i) {
            float4 v = *(const float4*)(src + i * 4);
            *(float4*)&lds[k * SB + i * 4] = v;
        }
    }

    // async copy of one A chunk: 512 rows x 16 floats = 2048 b128 -> 8 per thread
    auto issue_chunk = [&](int kc, unsigned ldsbase) {
        #pragma unroll
        for (int it = 0; it < 8; ++it) {
            int idx = tid + it * 256;
            int row = idx >> 2;
            int q4  = idx & 3;
            unsigned laddr = ldsbase + (unsigned)(row * SA + q4 * 4) * 4u;
            unsigned voff  = (unsigned)(row * IN_CH + kc * KC + q4 * 4) * 4u;
            asm volatile("global_load_async_to_lds_b128 %0, %1, %2"
                         :: "v"(laddr), "v"(voff), "s"(xbase) : "memory");
        }
    };

    // ---- GEMM: 512x256 * 256x16, K in KC=16 chunks, double-buffered async ----
    v8f acc[4];
    #pragma unroll
    for (int t = 0; t < 4; ++t) acc[t] = (v8f){0.f,0.f,0.f,0.f,0.f,0.f,0.f,0.f};

    // one chunk of WMMA from a compile-time-constant LDS region (pure DS path)
    auto compute_chunk = [&](int kc, const int abase) {
        #pragma unroll
        for (int kk = 0; kk < KC; kk += 4) {
            // B fragment: v0 = row K=kk+2*half, v1 = K=kk+2*half+1, N=lr
            int kb = (kc * KC + kk + 2 * half) * SB + lr;
            v2f b;
            b.x = lds[kb];
            b.y = lds[kb + SB];
            #pragma unroll
            for (int t = 0; t < 4; ++t) {
                // A fragment: lanes 0-15 K=kk,kk+1 ; lanes 16-31 K=kk+2,kk+3 ; M=lr
                int ai = abase + (m0 + t * 16 + lr) * SA + kk + 2 * half;
                v2f a;
                a.x = lds[ai];
                a.y = lds[ai + 1];
                acc[t] = __builtin_amdgcn_wmma_f32_16x16x4_f32(
                    false, a, false, b, (short)0, acc[t], false, false);
            }
        }
    };

    issue_chunk(0, aoff0);
    for (int kc2 = 0; kc2 < NCHUNK; kc2 += 2) {
        // even chunk: compute from A0, prefetch kc2+1 into A1
        issue_chunk(kc2 + 1, aoff1);
        wait_asynccnt_8();                // chunk kc2 drained, kc2+1 in flight
        __syncthreads();
        compute_chunk(kc2, A0OFF);
        __syncthreads();                  // A0 free for next copy

        // odd chunk: compute from A1, prefetch kc2+2 into A0
        if (kc2 + 2 < NCHUNK) {
            issue_chunk(kc2 + 2, aoff0);
            wait_asynccnt_8();            // chunk kc2+1 drained, kc2+2 in flight
        } else {
            wait_asynccnt_0();
        }
        __syncthreads();
        compute_chunk(kc2 + 1, A1OFF);
        __syncthreads();                  // A1 free for next copy
    }

    // ---- scatter accumulators into sort buffer: sbuf[c*SS + row] ----
    #pragma unroll
    for (int t = 0; t < 4; ++t) {
        #pragma unroll
        for (int r = 0; r < 8; ++r) {
            int row = m0 + t * 16 + r + 8 * half;   // D layout: VGPR r -> M=r / r+8
            lds[SBOFF + lr * SS + row] = acc[t][r]; // N = lr
        }
    }
    __syncthreads();

    // ---- bitonic sort: 16 independent ascending sorts of 512 (in-place LDS) ----
    for (int k = 2; k <= NPG; k <<= 1) {
        for (int j = k >> 1; j > 0; j >>= 1) {
            #pragma unroll
            for (int it = 0; it < 16; ++it) {
                int p   = tid + it * 256;           // 4096 pairs per stage
                int col = p >> 8;
                int t   = p & 255;
                int i   = ((t & ~(j - 1)) << 1) | (t & (j - 1));
                int ip  = i + j;
                bool asc = ((i & k) == 0);
                int bi  = SBOFF + col * SS;
                float a = lds[bi + i];
                float b = lds[bi + ip];
                float lo = fminf(a, b);
                float hi = fmaxf(a, b);
                lds[bi + i]  = asc ? lo : hi;
                lds[bi + ip] = asc ? hi : lo;
            }
            __syncthreads();
        }
    }

    // ---- quantile gather + scale, coalesced 64B per thread ----
    {
        const int q = tid;                          // 256 quantiles
        const int iq = (int)floorf(cw[q] * (float)(NPG - 1));
        const float inv_scale = 1.0f / 256.0f;      // (256*256)^(1/2)
        float* dst = out + (size_t)g * (NUM_QUANT * NUM_PROJ) + q * NUM_PROJ + pt * 16;
        #pragma unroll
        for (int i = 0; i < 4; ++i) {
            float4 v;
            v.x = lds[SBOFF + (i * 4 + 0) * SS + iq] * inv_scale;
            v.y = lds[SBOFF + (i * 4 + 1) * SS + iq] * inv_scale;
            v.z = lds[SBOFF + (i * 4 + 2) * SS + iq] * inv_scale;
            v.w = lds[SBOFF + (i * 4 + 3) * SS + iq] * inv_scale;
            *(float4*)(dst + i * 4) = v;
        }
    }
}

extern "C" void kernel_launch(void* const* d_in, const int* in_sizes, int n_in,
                              void* d_out, int out_size, void* d_ws, size_t ws_size,
                              hipStream_t stream) {
    const float* x    = (const float*)d_in[0];
    const float* proj = (const float*)d_in[1];
    const float* cw   = (const float*)d_in[2];
    // d_in[3] = batch (contiguous, implied), d_in[4] = num_graphs (256) -- unused
    float* out = (float*)d_out;

    dim3 grid(NUM_PROJ / 16, NUM_GRAPHS);
    dim3 block(256);
    swe_fused_kernel<<<grid, block, 0, stream>>>(x, proj, cw, out);
}